// HOGLayer_83107617178289
// MI455X (gfx1250) — compile-verified
//
#include <hip/hip_runtime.h>
#include <hip/hip_bf16.h>

// ---------------------------------------------------------------------------
// HOG for MI455X (gfx1250). Memory-bound (~260-300 MB HBM traffic ~= 12 us at
// 23.3 TB/s). No matrix contraction -> no WMMA; the CDNA5 feature used is the
// Tensor Data Mover: double-buffered tensor_load_to_lds of RGB strips into
// LDS, synchronized with s_wait_tensorcnt, overlapping DMA with compute.
// ---------------------------------------------------------------------------

typedef __attribute__((ext_vector_type(4))) unsigned int v4u;
typedef __attribute__((ext_vector_type(8))) int          v8i;
typedef __attribute__((ext_vector_type(4))) int          v4i;

#define IMG_H   512
#define IMG_W   512
#define NBATCH  64
#define PR      8       // pixels per cell (each dim)
#define NCR     64      // cell rows
#define NCC     64      // cell cols
#define OBINS   9
#define XTILE   128     // pixels of x handled per workgroup
#define CRG     8       // cell rows per workgroup
#define THREADS 256

// -------------------------------- kernel 1 ---------------------------------
// One WG per (batch, group-of-8 cell rows, 128px x-tile). Loops over 8 strips
// of 10 rows x (<=130) cols x 3 ch, TDM-loaded into double-buffered LDS.
__global__ __launch_bounds__(THREADS)
void hog_hist_kernel(const float* __restrict__ x, float* __restrict__ hist_g)
{
    __shared__ float rgb[2][10 * 390];   // packed [nrows][ncols*3], pitch=ncols*3
    __shared__ float gray[10 * 130];     // packed [nrows][ncols]
    __shared__ float hist[16 * OBINS];   // 16 cells x 9 bins

    const int wg = blockIdx.x;           // 64 * 8 * 4 = 2048
    const int xt = wg & 3;
    const int sg = (wg >> 2) & 7;
    const int b  = wg >> 5;

    const int X0       = xt * XTILE;
    const int gx0      = (X0 == 0) ? 0 : X0 - 1;                       // left halo
    const int gx1      = (X0 + XTILE >= IMG_W) ? (IMG_W - 1) : (X0 + XTILE);
    const int ncols    = gx1 - gx0 + 1;                                // 129 or 130
    const int pitch    = ncols * 3;                                    // rgb floats/row
    const int colshift = X0 - gx0;                                     // 0 or 1

    const int  tid      = threadIdx.x;
    const bool is_wave0 = (tid < 32);

    // Issue a TDM 2D tile load of strip s into buffer bb (uniform descriptor).
    auto issue = [&](int s, int bb) {
        const int cr    = sg * CRG + s;
        const int gy0   = (cr == 0) ? 0 : cr * PR - 1;
        const int gy1   = (cr == NCR - 1) ? (IMG_H - 1) : cr * PR + PR;
        const int nrows = gy1 - gy0 + 1;                               // 9 or 10
        unsigned long long ga = (unsigned long long)(const void*)x
            + (unsigned long long)((b * IMG_H + gy0) * IMG_W + gx0) * 12ull;
        unsigned lds = (unsigned)(unsigned long long)(const void*)&rgb[bb][0];
        v4u g0;
        g0[0] = 1u;                                     // count=1 (valid D#)
        g0[1] = lds;                                    // lds_addr
        g0[2] = (unsigned)ga;                           // global_addr[31:0]
        g0[3] = ((unsigned)(ga >> 32) & 0x01FFFFFFu)    // global_addr[56:32]
              | 0x80000000u;                            // type=2 ("image")
        const unsigned td0 = (unsigned)pitch;           // tile width, elements
        v8i g1;
        g1[0] = 0x00020000;                             // data_size=4B
        g1[1] = (int)(td0 << 16);                       // tensor_dim0[15:0]
        g1[2] = (int)((unsigned)nrows << 16);           // tensor_dim1[15:0]
        g1[3] = (int)(td0 << 16);                       // tile_dim0
        g1[4] = nrows;                                  // tile_dim1 (tile_dim2=0)
        g1[5] = IMG_W * 3;                              // tensor_dim0_stride
        g1[6] = 0;
        g1[7] = 0;
        v4i z4 = {0, 0, 0, 0};                          // 2D: groups 2/3 unused
        v8i z8 = {0, 0, 0, 0, 0, 0, 0, 0};
        __builtin_amdgcn_tensor_load_to_lds(g0, g1, z4, z4, z8, 0);
    };

    if (is_wave0) issue(0, 0);

    for (int s = 0; s < CRG; ++s) {
        const int cr     = sg * CRG + s;
        const int gy0    = (cr == 0) ? 0 : cr * PR - 1;
        const int gy1    = (cr == NCR - 1) ? (IMG_H - 1) : cr * PR + PR;
        const int nrows  = gy1 - gy0 + 1;
        const int rowoff = cr * PR - gy0;               // lds row of pixel row 0
        const int bb     = s & 1;

        if (is_wave0) {
            if (s + 1 < CRG) {                          // prefetch next strip
                issue(s + 1, (s + 1) & 1);
                __builtin_amdgcn_s_wait_tensorcnt(1);   // strip s done (in-order)
            } else {
                __builtin_amdgcn_s_wait_tensorcnt(0);
            }
        }
        __syncthreads();   // strip s resident; previous iteration fully retired

        // --- RGB -> gray (LDS -> LDS) ---
        const int ntex = nrows * ncols;
        for (int i = tid; i < ntex; i += THREADS) {
            const int rr = i / ncols;
            const int cc = i - rr * ncols;
            const int p  = (rr * ncols + cc) * 3;
            gray[rr * ncols + cc] = 0.2125f * rgb[bb][p + 0]
                                  + 0.7154f * rgb[bb][p + 1]
                                  + 0.0721f * rgb[bb][p + 2];
        }
        if (tid < 16 * OBINS) hist[tid] = 0.0f;
        __syncthreads();

        // --- gradients + orientation histogram: 16 threads/cell, 4 px each ---
        {
            const int cell = tid >> 4;            // 0..15
            const int sub  = tid & 15;
            const int r    = sub >> 1;            // pixel row in cell (0..7)
            const int xq   = (sub & 1) * 4;       // 4-px column quad
            const int y    = cr * PR + r;
            const int lr   = rowoff + r;          // lds row of pixel
            float acc[OBINS];
#pragma unroll
            for (int o = 0; o < OBINS; ++o) acc[o] = 0.0f;
#pragma unroll
            for (int k = 0; k < 4; ++k) {
                const int lx = cell * 8 + xq + k; // 0..127 in tile
                const int gx = X0 + lx;
                const int lc = lx + colshift;
                float gr = 0.0f, gc = 0.0f;
                if (y > 0 && y < IMG_H - 1)
                    gr = gray[(lr + 1) * ncols + lc] - gray[(lr - 1) * ncols + lc];
                if (gx > 0 && gx < IMG_W - 1)
                    gc = gray[lr * ncols + lc + 1] - gray[lr * ncols + lc - 1];
                const float mag = sqrtf(gr * gr + gc * gc);
                float ang = atan2f(gr, gc) * 57.295779513082321f;  // rad2deg
                if (ang < 0.0f)    ang += 180.0f;
                if (ang >= 180.0f) ang -= 180.0f;                  // [0,180)
                int bin = (int)(ang * 0.05f);                      // /20 deg
                bin = bin < 0 ? 0 : (bin > OBINS - 1 ? OBINS - 1 : bin);
                acc[bin] += mag;
            }
#pragma unroll
            for (int o = 0; o < OBINS; ++o)
                atomicAdd(&hist[cell * OBINS + o], acc[o]);        // ds_add_f32
        }
        __syncthreads();

        // --- write cell histograms (divided by cell area 64) ---
        if (tid < 16 * OBINS) {
            const int cell = tid / OBINS;
            const int o    = tid - cell * OBINS;
            const int cc   = xt * 16 + cell;
            hist_g[((b * NCR + cr) * NCC + cc) * OBINS + o] =
                hist[tid] * (1.0f / 64.0f);
        }
        // next iteration's leading __syncthreads covers write-vs-reuse hazards
    }
}

// -------------------------------- kernel 2 ---------------------------------
// One thread per 2x2-cell block: norm -> clip(0.2) -> renorm, 36 outputs.
__global__ __launch_bounds__(THREADS)
void hog_norm_kernel(const float* __restrict__ hist_g, float* __restrict__ out)
{
    const int NBLK = 63 * 63;
    const int tid  = blockIdx.x * blockDim.x + threadIdx.x;
    if (tid >= NBATCH * NBLK) return;
    const int b   = tid / NBLK;
    const int rem = tid - b * NBLK;
    const int rb  = rem / 63;
    const int cb  = rem - rb * 63;

    float v[36];
    float s = 0.0f;
#pragma unroll
    for (int r = 0; r < 2; ++r)
#pragma unroll
        for (int c = 0; c < 2; ++c) {
            const float* hp = &hist_g[((b * NCR + rb + r) * NCC + (cb + c)) * OBINS];
#pragma unroll
            for (int o = 0; o < OBINS; ++o) {
                const float h = hp[o];
                v[(r * 2 + c) * OBINS + o] = h;
                s += h * h;
            }
        }
    const float rn = rsqrtf(s + 1e-10f);   // sqrt(sum + EPS^2)
    float s2 = 0.0f;
#pragma unroll
    for (int i = 0; i < 36; ++i) {
        float t = fminf(v[i] * rn, 0.2f);
        v[i] = t;
        s2 += t * t;
    }
    const float rn2 = rsqrtf(s2 + 1e-10f);
    float* op = &out[(size_t)tid * 36];
#pragma unroll
    for (int i = 0; i < 36; ++i) op[i] = v[i] * rn2;
}

// ------------------------------ launcher -----------------------------------
extern "C" void kernel_launch(void* const* d_in, const int* in_sizes, int n_in,
                              void* d_out, int out_size, void* d_ws, size_t ws_size,
                              hipStream_t stream)
{
    const float* x    = (const float*)d_in[0];
    float*       out  = (float*)d_out;
    float*       hist = (float*)d_ws;     // 64*64*64*9 floats = 9.44 MB

    hog_hist_kernel<<<dim3(NBATCH * 8 * 4), dim3(THREADS), 0, stream>>>(x, hist);

    const int n2 = NBATCH * 63 * 63;
    hog_norm_kernel<<<dim3((n2 + THREADS - 1) / THREADS), dim3(THREADS), 0, stream>>>(hist, out);
}